// LogisticRegression_52845277610636
// MI455X (gfx1250) — compile-verified
//
#include <hip/hip_runtime.h>
#include <hip/hip_bf16.h>
#include <math.h>

#define VOCAB 50000
#define EMB   300
#define BATCH 1024
#define SEQ   200

typedef float v2f __attribute__((ext_vector_type(2)));
typedef float v8f __attribute__((ext_vector_type(8)));

// ---------------------------------------------------------------------------
// Kernel 1: ew[v] = dot(E[v, 0:300], W[0:300]) for all v, via
// V_WMMA_F32_16X16X4_F32. One wave32 per 16-row vocab tile.
//   A (16x4 f32): lane&15 = row within tile; lanes 0-15 carry K=k,k+1,
//                 lanes 16-31 carry K=k+2,k+3 (ISA 32-bit A layout).
//   B (4x16 f32): only column N=0 carries W1; built from a *uniform*
//                 s_load of W[k..k+3] + v_cndmask (no divergent loads).
//   D column N=0 (lanes 0 and 16) holds the 16 accumulated dot products.
// Grid is exact: 3125 tiles = 625 blocks x 5 waves -> no bounds check.
// Memory-bound: streams the 60 MB embedding table once (~2.7 us @ 23.3 TB/s).
// ---------------------------------------------------------------------------
__global__ __launch_bounds__(160) void ew_gemv_wmma(
    const float* __restrict__ E, const float* __restrict__ W,
    float* __restrict__ ew) {
  const int lane = threadIdx.x & 31;
  const int wave = threadIdx.x >> 5;
  const int tile = blockIdx.x * 5 + wave;       // 625 * 5 == 3125 exactly
  const int rowbase = tile * 16;
  const int g = lane >> 4;                      // half-wave: K offset 0 or 2
  const bool n0 = (lane & 15) == 0;             // matrix column N=0 carries W1
  const float* Erow = E + (size_t)(rowbase + (lane & 15)) * EMB;

  v8f c = {};
  for (int k = 0; k < EMB; k += 4) {            // 75 WMMA ops per tile
    const int kk = k + 2 * g;
    v2f a;
    a.x = Erow[kk];                             // contiguous pair -> b64 load
    a.y = Erow[kk + 1];

    // Uniform address (k uniform, W kernarg) -> scalar s_load_b128.
    const float4 wv = *(const float4*)(W + k);
    const float w0 = g ? wv.z : wv.x;           // v_cndmask, no branch
    const float w1 = g ? wv.w : wv.y;
    v2f bm;
    bm.x = n0 ? w0 : 0.0f;
    bm.y = n0 ? w1 : 0.0f;

    // 8 args: (neg_a, A, neg_b, B, c_mod, C, reuse_a, reuse_b)
    c = __builtin_amdgcn_wmma_f32_16x16x4_f32(false, a, false, bm,
                                              (short)0, c, false, false);
  }

  // D layout (32-bit C/D 16x16): VGPR r -> M=r (lanes 0-15), M=8+r (16-31).
  // Column N=0 lives in lane 0 (M=0..7) and lane 16 (M=8..15).
  if (n0) {
    float* dst = ew + rowbase + g * 8;
#pragma unroll
    for (int r = 0; r < 8; ++r) dst[r] = c[r];
  }
}

// ---------------------------------------------------------------------------
// Kernel 2: per-row logit. One wave32 per batch row (8 rows / 256-thr block).
//   acc = (1/L) * sum_j ew[x[i,j]]  +  sum_{v in unique(x[i,:])} W2[v]
// Dedup matches jnp .at[].set(1.0): duplicates count once (token contributes
// W2 only if no earlier position holds the same index). Row cached in LDS,
// wave32 shuffle reduction, sigmoid at the end. Traffic ~1.6 MB total.
// ---------------------------------------------------------------------------
__global__ __launch_bounds__(256) void row_logit(
    const int* __restrict__ x, const float* __restrict__ W,
    const float* __restrict__ bscal, const float* __restrict__ ew,
    float* __restrict__ out) {
  __shared__ int sx[8][SEQ];
  const int lane = threadIdx.x & 31;
  const int wave = threadIdx.x >> 5;
  const int row = blockIdx.x * 8 + wave;
  const int* xr = x + (size_t)row * SEQ;

  for (int j = lane; j < SEQ; j += 32) sx[wave][j] = xr[j];
  __syncthreads();

  float acc = 0.0f;
  for (int j = lane; j < SEQ; j += 32) {
    const int v = sx[wave][j];
    acc += ew[v] * (1.0f / (float)SEQ);
    bool dup = false;
    for (int jj = 0; jj < j; ++jj) {
      if (sx[wave][jj] == v) { dup = true; break; }
    }
    if (!dup) acc += W[EMB + v];
  }

  // wave32 tree reduction
  for (int off = 16; off > 0; off >>= 1)
    acc += __shfl_down(acc, off, 32);

  if (lane == 0) {
    const float z = acc + bscal[0];
    out[row] = 1.0f / (1.0f + __expf(-z));
  }
}

// ---------------------------------------------------------------------------
// Launch. Inputs (setup_inputs order): x(int32 [1024,200]),
// embedding_weight(f32 [50000,300]), W(f32 [1,50300]), b(f32 [1]).
// Output: 1024 f32. Workspace: ew table, 50000 f32 = 200 KB, fully
// written by kernel 1 before kernel 2 reads it (same stream, in order).
// ---------------------------------------------------------------------------
extern "C" void kernel_launch(void* const* d_in, const int* in_sizes, int n_in,
                              void* d_out, int out_size, void* d_ws, size_t ws_size,
                              hipStream_t stream) {
  const int*   x = (const int*)d_in[0];
  const float* E = (const float*)d_in[1];
  const float* W = (const float*)d_in[2];
  const float* b = (const float*)d_in[3];
  float* out = (float*)d_out;
  float* ew  = (float*)d_ws;

  // 3125 sixteen-row tiles = 625 blocks x 5 wave32 (exact, no tail guard)
  ew_gemv_wmma<<<625, 160, 0, stream>>>(E, W, ew);

  row_logit<<<BATCH / 8, 256, 0, stream>>>(x, W, b, ew, out);
}